// AdvancedWaveletDecomp_80985903333556
// MI455X (gfx1250) — compile-verified
//
#include <hip/hip_runtime.h>
#include <hip/hip_bf16.h>

#define BB 32
#define CC 128
#define LL 4096
#define FLT 9
#define DIMM 512
#define NN (BB*CC*LL)   // 16777216

typedef __attribute__((ext_vector_type(2))) float v2f;
typedef __attribute__((ext_vector_type(8))) float v8f;

__device__ __forceinline__ float gelu_f(float x){ return 0.5f*x*(1.0f+erff(x*0.70710678118654752f)); }
__device__ __forceinline__ float sigm_f(float x){ return 1.0f/(1.0f+expf(-x)); }

// ---------------- gates_w transpose: [lvl][m][ci][k] -> [lvl][m][k][ci] ----------------
// Makes WMMA A-fragment pairs (ci, ci+1) contiguous -> global_load_b64 path.
__global__ __launch_bounds__(256) void k_wt(const float* __restrict__ gw, float* __restrict__ wt){
    int o = blockIdx.x*256 + threadIdx.x;        // o = ((lvl*CC+m)*3+k)*CC + ci ; total 3*CC*3*CC
    int ci = o % CC;
    int k  = (o/CC) % 3;
    int m  = (o/(CC*3)) % CC;
    int lvl= o/(CC*3*CC);
    wt[o] = gw[(size_t)((lvl*CC + m)*CC + ci)*3 + k];
}

// ---------------- feat = mean over L ----------------
__global__ __launch_bounds__(256) void k_mean(const float* __restrict__ src, float* __restrict__ feat){
    __shared__ float red[256];
    int row = blockIdx.x; int t = threadIdx.x;
    const float4* p4 = (const float4*)(src + (size_t)row*LL);
    float s = 0.f;
    #pragma unroll
    for(int i=0;i<LL/1024;i++){ float4 v = p4[t + i*256]; s += (v.x+v.y)+(v.z+v.w); }
    red[t] = s; __syncthreads();
    for(int off=128; off>0; off>>=1){ if(t<off) red[t]+=red[t+off]; __syncthreads(); }
    if(t==0) feat[row] = red[0] * (1.0f/LL);
}

// ---------------- fused filter MLP (one block per batch row) ----------------
__global__ __launch_bounds__(512) void k_mlp(const float* __restrict__ feat,
        const float* __restrict__ stat_w, const float* __restrict__ stat_b,
        const float* __restrict__ wg1_w,  const float* __restrict__ wg1_b,
        const float* __restrict__ wg2_w,  const float* __restrict__ wg2_b,
        float* __restrict__ filt, float* __restrict__ lo_all, float* __restrict__ hi_all, int lvl){
    __shared__ float sf[CC];
    __shared__ float h1[DIMM];
    __shared__ float h2[2*DIMM];
    int b = blockIdx.x, t = threadIdx.x;
    if(t<CC) sf[t] = feat[b*CC+t];
    __syncthreads();
    { float s = stat_b[t]; const float* w = stat_w + (size_t)t*CC;
      for(int c=0;c<CC;c++) s += w[c]*sf[c];
      h1[t] = gelu_f(s); }
    __syncthreads();
    for(int j=t;j<2*DIMM;j+=DIMM){
        float s = wg1_b[j]; const float* w = wg1_w + (size_t)j*DIMM;
        for(int d=0;d<DIMM;d++) s += w[d]*h1[d];
        h2[j] = gelu_f(s);
    }
    __syncthreads();
    if(t<2*FLT){
        float s = wg2_b[t]; const float* w = wg2_w + (size_t)t*(2*DIMM);
        for(int d=0;d<2*DIMM;d++) s += w[d]*h2[d];
        filt[b*(2*FLT)+t] = s;
        if(t<FLT) lo_all[lvl*(BB*FLT) + b*FLT + t]       = s;
        else      hi_all[lvl*(BB*FLT) + b*FLT + (t-FLT)] = s;
    }
}

// ---------------- ortho loss (last level only) + energy=0 ----------------
__global__ __launch_bounds__(64) void k_ortho(const float* __restrict__ filt, float* __restrict__ scal){
    __shared__ float sS[32], sA[32], sM[32];
    int t = threadIdx.x;
    if(t<32){
        float lo[FLT]; float n2=0.f;
        #pragma unroll
        for(int k=0;k<FLT;k++){ lo[k]=filt[t*(2*FLT)+k]; n2 += lo[k]*lo[k]; }
        float denom = sqrtf(n2) + 1e-8f;
        float S=0.f;
        #pragma unroll
        for(int k=0;k<FLT;k++) S += fabsf(lo[k]/denom);
        float sum2 = n2/(denom*denom);
        float sm = fabsf(lo[0]) + fabsf(lo[FLT-1]);
        #pragma unroll
        for(int k=1;k<FLT;k++) sm += fabsf(lo[k]-lo[k-1]);
        sS[t]=S*S; sA[t]=fabsf(sum2-1.0f); sM[t]=sm;
    }
    __syncthreads();
    if(t==0){
        float aS=0,aA=0,aM=0;
        for(int i=0;i<32;i++){aS+=sS[i];aA+=sA[i];aM+=sM[i];}
        float shift = 3.0f * (aS/32.0f) * (1.0f/81.0f);   // roll-invariant row-abs-sum identity
        float amp   = aA/32.0f;
        float smooth= aM * (1.0f/320.0f);
        scal[0] = 0.01f*(shift+amp) + 0.1f*smooth;
        scal[1] = 0.0f;  // energy_loss (dx_dt == 0)
    }
}

// ---------------- per-sample depthwise 9-tap, edge padded (float4 path) ----------------
__global__ __launch_bounds__(256) void k_dwt(const float* __restrict__ src, const float* __restrict__ filt,
                                             float* __restrict__ outA, float* __restrict__ outD){
    __shared__ float tile[1032];           // cols l0-4 .. l0+1027
    int blk = blockIdx.x;
    int row = blk >> 2;                    // b*C + c
    int l0  = (blk & 3) << 10;
    int t = threadIdx.x;
    int b = row >> 7;
    const float* p = src + (size_t)row*LL;
    *(float4*)&tile[4 + 4*t] = *(const float4*)(p + l0 + 4*t);
    if(t < 4){ int g = max(l0 - 4 + t, 0); tile[t] = p[g]; }
    else if(t < 8){ int g = min(l0 + 1024 + (t-4), LL-1); tile[1028 + (t-4)] = p[g]; }
    __syncthreads();
    float lo[FLT], hi[FLT];
    #pragma unroll
    for(int k=0;k<FLT;k++){ lo[k]=filt[b*(2*FLT)+k]; hi[k]=filt[b*(2*FLT)+FLT+k]; }
    float aa[4], dd[4];
    #pragma unroll
    for(int i=0;i<4;i++){
        float sa=0.f, sd=0.f;
        #pragma unroll
        for(int k=0;k<FLT;k++){ float v = tile[4*t + i + k]; sa += lo[k]*v; sd += hi[k]*v; }
        aa[i]=sa; dd[i]=sd;
    }
    size_t o = (size_t)row*LL + l0 + 4*t;
    *(float4*)(outA + o) = make_float4(aa[0],aa[1],aa[2],aa[3]);
    *(float4*)(outD + o) = make_float4(dd[0],dd[1],dd[2],dd[3]);
}

// ---------------- WMMA attn 1x1 conv C->C/4, gelu ----------------
// wave w: m-tile (w>>2)*16, n-subtile (w&3)*16 ; K=128 -> 32 wmma steps
__global__ __launch_bounds__(256) void k_attn1(const float* __restrict__ cur, const float* __restrict__ w1,
                                               const float* __restrict__ b1, float* __restrict__ T){
    __shared__ float X[CC*72];
    int b  = blockIdx.x >> 6;
    int l0 = (blockIdx.x & 63) << 6;
    int t = threadIdx.x;
    const float* base = cur + (size_t)b*CC*LL;
    for(int idx=t; idx<CC*16; idx+=256){
        int r = idx >> 4, j = idx & 15;
        *(float4*)&X[r*72 + 4*j] = *(const float4*)(base + (size_t)r*LL + l0 + 4*j);
    }
    __syncthreads();
    int wv = t>>5, lane = t&31, half = lane>>4, lm = lane&15;
    int m0 = (wv>>2)<<4;          // 0 or 16
    int ns = wv&3;
    v8f acc = {};
    const float* wr = w1 + (size_t)(m0+lm)*CC;
    for(int ck=0; ck<CC; ck+=4){
        int ci = ck + 2*half;
        v2f a;  a.x = wr[ci];  a.y = wr[ci+1];           // contiguous pair
        int bb = ci*72 + ns*16 + lm;
        v2f bf; bf.x = X[bb];  bf.y = X[bb+72];
        acc = __builtin_amdgcn_wmma_f32_16x16x4_f32(false, a, false, bf, (short)0, acc, false, false);
    }
    #pragma unroll
    for(int v=0; v<8; v++){
        int j = m0 + v + 8*half;
        int l = l0 + ns*16 + lm;
        T[(size_t)b*32*LL + (size_t)j*LL + l] = gelu_f(acc[v] + b1[j]);
    }
}

// ---------------- WMMA attn 1x1 conv C/4->C, sigmoid, yh *= (1+attn) in place ----------------
__global__ __launch_bounds__(256) void k_attn2(const float* __restrict__ T, const float* __restrict__ w2,
                                               const float* __restrict__ b2, float* __restrict__ yh){
    __shared__ float Xt[32*72];
    int b  = blockIdx.x >> 6;
    int l0 = (blockIdx.x & 63) << 6;
    int t = threadIdx.x;
    const float* base = T + (size_t)b*32*LL;
    for(int idx=t; idx<32*16; idx+=256){
        int r = idx >> 4, j = idx & 15;
        *(float4*)&Xt[r*72 + 4*j] = *(const float4*)(base + (size_t)r*LL + l0 + 4*j);
    }
    __syncthreads();
    int wv = t>>5, lane = t&31, half = lane>>4, lm = lane&15;
    int m0 = wv<<4;
    v8f acc[4] = {{},{},{},{}};
    const float* wr = w2 + (size_t)(m0+lm)*32;
    for(int ck=0; ck<32; ck+=4){
        int ci = ck + 2*half;
        v2f a;  a.x = wr[ci];  a.y = wr[ci+1];
        int bb = ci*72 + lm;
        #pragma unroll
        for(int ns=0; ns<4; ns++){
            v2f bf; bf.x = Xt[bb + ns*16]; bf.y = Xt[bb + ns*16 + 72];
            acc[ns] = __builtin_amdgcn_wmma_f32_16x16x4_f32(false, a, false, bf, (short)0, acc[ns], false, false);
        }
    }
    #pragma unroll
    for(int ns=0; ns<4; ns++){
        #pragma unroll
        for(int v=0; v<8; v++){
            int c = m0 + v + 8*half;
            int pos = ns*16 + lm;
            size_t gi = (size_t)b*CC*LL + (size_t)c*LL + l0 + pos;
            yh[gi] = yh[gi]*(1.0f + sigm_f(acc[ns][v] + b2[c]));
        }
    }
}

// ---------------- WMMA gate conv (3-tap C->C GEMM) + fused sigmoid-gate update ----------------
// out[b,c,l] = cur[b,c,l] + sigmoid(conv3(cur)[b,c,l] + bias[c]) * detail[b,c,l]
// Wt layout: [c_out][tap][c_in]  (tap-major -> contiguous A pairs)
__global__ __launch_bounds__(256) void k_gate_wmma(const float* __restrict__ cur, const float* __restrict__ detail,
                                                   float* __restrict__ outp, const float* __restrict__ Wt,
                                                   const float* __restrict__ bias){
    __shared__ float X[CC*72];   // body cols 4..67 <-> l0..l0+63 ; halo col 3 (l0-1) and 68 (l0+64)
    int b  = blockIdx.x >> 6;
    int l0 = (blockIdx.x & 63) << 6;
    int t = threadIdx.x;
    const float* base = cur + (size_t)b*CC*LL;
    for(int idx=t; idx<CC*16; idx+=256){        // vectorized body fill
        int r = idx >> 4, j = idx & 15;
        *(float4*)&X[r*72 + 4 + 4*j] = *(const float4*)(base + (size_t)r*LL + l0 + 4*j);
    }
    {   // halo: one element per thread (128 left + 128 right)
        int r = t & 127;
        if(t < 128){
            int gl = l0 - 1;
            X[r*72 + 3]  = (gl >= 0) ? base[(size_t)r*LL + gl] : 0.f;
        } else {
            int gl = l0 + 64;
            X[r*72 + 68] = (gl < LL) ? base[(size_t)r*LL + gl] : 0.f;
        }
    }
    __syncthreads();

    int wv   = t >> 5;        // wave 0..7 -> c_out tile
    int lane = t & 31;
    int half = lane >> 4;     // 0/1
    int lm   = lane & 15;
    int m0   = wv << 4;

    v8f acc[4] = {{},{},{},{}};
    const float* wr = Wt + (size_t)(m0+lm)*CC*3;  // this lane's output-channel row [tap][c_in]

    for(int k=0;k<3;k++){
        for(int ck=0; ck<CC; ck+=4){
            int ci = ck + 2*half;
            v2f a;                                   // A 16x4: m=lm, k-pair by lane half
            a.x = wr[k*CC + ci];                     // contiguous pair -> b64
            a.y = wr[k*CC + ci + 1];
            int bb = ci*72 + lm + k + 3;             // B 4x16: n=lm, k-pair by lane half
            #pragma unroll
            for(int ns=0; ns<4; ns++){
                v2f bf;
                bf.x = X[bb + ns*16];
                bf.y = X[bb + ns*16 + 72];
                acc[ns] = __builtin_amdgcn_wmma_f32_16x16x4_f32(
                    false, a, false, bf, (short)0, acc[ns], false, false);
            }
        }
    }

    // epilogue: C/D layout c = m0 + v + 8*half, l = l0 + ns*16 + lm
    #pragma unroll
    for(int ns=0; ns<4; ns++){
        #pragma unroll
        for(int v=0; v<8; v++){
            int c = m0 + v + 8*half;
            int pos = ns*16 + lm;
            float g  = sigm_f(acc[ns][v] + bias[c]);
            float cv = X[c*72 + pos + 4];            // cur[b,c,l] from LDS
            size_t gi = (size_t)b*CC*LL + (size_t)c*LL + l0 + pos;
            outp[gi] = cv + g * detail[gi];
        }
    }
}

extern "C" void kernel_launch(void* const* d_in, const int* in_sizes, int n_in,
                              void* d_out, int out_size, void* d_ws, size_t ws_size,
                              hipStream_t stream){
    const float* x      = (const float*)d_in[0];
    const float* stat_w = (const float*)d_in[1];
    const float* stat_b = (const float*)d_in[2];
    const float* wg1_w  = (const float*)d_in[3];
    const float* wg1_b  = (const float*)d_in[4];
    const float* wg2_w  = (const float*)d_in[5];
    const float* wg2_b  = (const float*)d_in[6];
    const float* gates_w= (const float*)d_in[7];
    const float* gates_b= (const float*)d_in[8];
    const float* a1_w   = (const float*)d_in[9];
    const float* a1_b   = (const float*)d_in[10];
    const float* a2_w   = (const float*)d_in[11];
    const float* a2_b   = (const float*)d_in[12];

    float* out = (float*)d_out;
    float* yl  = out;
    float* yh0 = out + (size_t)NN;
    float* yh1 = out + 2*(size_t)NN;
    float* yh2 = out + 3*(size_t)NN;
    float* scal   = out + 4*(size_t)NN;          // [ortho, energy]
    float* lo_all = scal + 2;                    // (3,32,9)
    float* hi_all = lo_all + 3*BB*FLT;

    float* ws   = (float*)d_ws;
    float* A    = ws;                            // N floats (ping-pong with yl region)
    float* T1   = ws + (size_t)NN;               // B * C/4 * L
    float* feat = T1 + (size_t)BB*32*LL;         // 32*128
    float* filt = feat + BB*CC;                  // 32*18
    float* Wt   = filt + BB*2*FLT;               // 3*128*3*128 transposed gate weights

    const int gridL = BB*(LL/64);                // 2048

    // ---- transpose gate weights (tap-major) ----
    k_wt<<<(3*CC*3*CC)/256, 256, 0, stream>>>(gates_w, Wt);

    // ---- decomposition: x -> A -> yl -> A  (details straight into yh regions) ----
    const float* srcs[3] = { x, A, yl };
    float*       dsts[3] = { A, yl, A };
    float*       dets[3] = { yh0, yh1, yh2 };
    for(int lvl=0; lvl<3; lvl++){
        k_mean<<<BB*CC, 256, 0, stream>>>(srcs[lvl], feat);
        k_mlp<<<BB, DIMM, 0, stream>>>(feat, stat_w, stat_b, wg1_w, wg1_b, wg2_w, wg2_b,
                                       filt, lo_all, hi_all, lvl);
        if(lvl==2) k_ortho<<<1, 64, 0, stream>>>(filt, scal);
        k_dwt<<<BB*CC*(LL/1024), 256, 0, stream>>>(srcs[lvl], filt, dsts[lvl], dets[lvl]);
    }

    // ---- reconstruction: currents A -> yl -> A -> yl ----
    // i = 2 (no attn)
    k_gate_wmma<<<gridL, 256, 0, stream>>>(A, yh2, yl, Wt + 2*CC*CC*3, gates_b + 2*CC);
    // i = 1
    k_attn1<<<gridL, 256, 0, stream>>>(yl, a1_w + 1*32*CC, a1_b + 1*32, T1);
    k_attn2<<<gridL, 256, 0, stream>>>(T1, a2_w + 1*CC*32, a2_b + 1*CC, yh1);
    k_gate_wmma<<<gridL, 256, 0, stream>>>(yl, yh1, A, Wt + 1*CC*CC*3, gates_b + 1*CC);
    // i = 0
    k_attn1<<<gridL, 256, 0, stream>>>(A, a1_w, a1_b, T1);
    k_attn2<<<gridL, 256, 0, stream>>>(T1, a2_w, a2_b, yh0);
    k_gate_wmma<<<gridL, 256, 0, stream>>>(A, yh0, yl, Wt, gates_b);
}